// GroupedEmbedding_49864570306745
// MI455X (gfx1250) — compile-verified
//
#include <hip/hip_runtime.h>
#include <cstdint>

// Grouped embedding gather for MI455X (gfx1250).
// Pure bandwidth problem: ~1.65 GB moved -> ~70 us floor at 23.3 TB/s.
// One wave32 per 512-byte row (B128/lane) through the CDNA5 async global<->LDS
// copy engine (ASYNCcnt), double-buffered 10-row groups; stores of group G
// overlap gather-loads of group G+1, index fetches ride a group ahead.
// Work partition is exact: 16000 waves x 100 rows, each wave inside a single
// table -> no guards, no divergent control flow, scalar index/base math.

static constexpr unsigned kTables      = 8;
static constexpr unsigned kEmb         = 100000;
static constexpr unsigned kDim         = 128;
static constexpr unsigned kIds         = 200000;
static constexpr unsigned kRows        = kTables * kIds;   // 1,600,000
static constexpr unsigned kWavesPerBlk = 8;                // 256 thr / wave32
static constexpr unsigned kBlocks      = 2000;
static constexpr unsigned kNumWaves    = kBlocks * kWavesPerBlk;   // 16000
static constexpr unsigned kRowsPerWave = kRows / kNumWaves;        // 100 exact
static constexpr unsigned kGroup       = 10;               // rows per buffer
static constexpr unsigned kGroups      = kRowsPerWave / kGroup;    // 10 exact
static constexpr unsigned kRowBytes    = kDim * 4;         // 512 B

static_assert(kNumWaves * kRowsPerWave == kRows, "exact partition");
static_assert(kGroups * kGroup == kRowsPerWave, "exact groups");
static_assert(kIds % kRowsPerWave == 0, "wave never crosses a table boundary");

__global__ void __launch_bounds__(256)
grouped_embedding_gather(const int* __restrict__ values,
                         const float* __restrict__ weights,
                         float* __restrict__ out)
{
    // 8 waves * 2 buffers * 10 rows * 512B = 80 KB (of 320 KB WGP LDS)
    __shared__ __align__(16) float lds[kWavesPerBlk * 2 * kGroup * kDim];

    const unsigned lane   = threadIdx.x & 31u;
    const unsigned wib    = threadIdx.x >> 5;
    const unsigned lane16 = lane * 16u;

    // Low 32 bits of a flat LDS pointer == byte offset within LDS.
    const unsigned lds0 = (unsigned)(uintptr_t)(&lds[0])
                        + wib * (2u * kGroup * kRowBytes)
                        + lane16;

    // Wave-uniform scalars: wave id, first row, table id, bases.
    const unsigned wid  = __builtin_amdgcn_readfirstlane(
                              blockIdx.x * kWavesPerBlk + wib);
    const unsigned row0 = wid * kRowsPerWave;          // contiguous 100-row span
    const unsigned t    = row0 / kIds;                 // whole wave in one table

    const int* vptr = values + row0;                   // uniform -> scalar loads
    const unsigned long long wtab =
        (unsigned long long)(uintptr_t)weights
        + (unsigned long long)(t * kEmb) * kRowBytes;  // table base (scalar)
    const unsigned long long obase =
        (unsigned long long)(uintptr_t)out
        + (unsigned long long)row0 * kRowBytes;        // output base (scalar)

    int idx[kGroup];

    auto fetch_indices = [&](unsigned G) {
#pragma unroll
        for (unsigned j = 0; j < kGroup; ++j)
            idx[j] = vptr[G * kGroup + j];             // uniform addr, no guards
    };

    auto issue_loads = [&](unsigned G) {
        const unsigned bufBase = lds0 + (G & 1u) * (kGroup * kRowBytes);
#pragma unroll
        for (unsigned j = 0; j < kGroup; ++j) {
            const unsigned long long ga =
                wtab + (unsigned)idx[j] * kRowBytes + lane16;
            const unsigned la = bufBase + j * kRowBytes;
            asm volatile("global_load_async_to_lds_b128 %0, %1, off"
                         :: "v"(la), "v"(ga) : "memory");
        }
    };

    auto issue_stores = [&](unsigned G) {
        const unsigned bufBase = lds0 + (G & 1u) * (kGroup * kRowBytes);
#pragma unroll
        for (unsigned j = 0; j < kGroup; ++j) {
            const unsigned long long ga =
                obase + (G * kGroup + j) * kRowBytes + lane16;
            const unsigned la = bufBase + j * kRowBytes;
            asm volatile("global_store_async_from_lds_b128 %0, %1, off th:TH_STORE_NT"
                         :: "v"(ga), "v"(la) : "memory");
        }
    };

    // Pipeline: each s_wait_asynccnt drains loads(G) + stores(G-1); between
    // drains, stores(G) and loads(G+1) are in flight together, and the index
    // fetch for G+2 overlaps the next drain.
    fetch_indices(0);
    issue_loads(0);
    fetch_indices(1);
    for (unsigned G = 0; G < kGroups; ++G) {
        asm volatile("s_wait_asynccnt 0x0" ::: "memory");
        issue_stores(G);
        if (G + 1 < kGroups) {
            issue_loads(G + 1);                        // consumes idx[] for G+1
            if (G + 2 < kGroups) fetch_indices(G + 2); // refill behind the wait
        }
    }
    asm volatile("s_wait_asynccnt 0x0" ::: "memory");
}

extern "C" void kernel_launch(void* const* d_in, const int* in_sizes, int n_in,
                              void* d_out, int out_size, void* d_ws, size_t ws_size,
                              hipStream_t stream) {
    (void)in_sizes; (void)n_in; (void)out_size; (void)d_ws; (void)ws_size;
    const int*   values  = (const int*)d_in[0];    // [8, 200000] indices
    const float* weights = (const float*)d_in[1];  // [8, 100000, 128] f32
    float*       out     = (float*)d_out;          // [1.6M, 128] f32

    hipLaunchKernelGGL(grouped_embedding_gather, dim3(kBlocks), dim3(256), 0, stream,
                       values, weights, out);
}